// SphericalBasisLayer_5119601016912
// MI455X (gfx1250) — compile-verified
//
#include <hip/hip_runtime.h>
#include <math.h>

#define NS 7
#define NR 6
#define NJ 42               // NS*NR
#define TILE_T 128
#define CUTOFF_INV 0.2f

struct Tables {
  float zeros[NJ];          // Bessel zeros z_{l,k}
  float norms[NJ];          // sqrt(2)/|j_{l+1}(z_{l,k})|
  float coefA[16][8];       // pref_l * monomial coeffs of P_l (rows >=7 zero)
};

typedef __attribute__((ext_vector_type(2))) float v2f;
typedef __attribute__((ext_vector_type(4))) float v4f;
typedef __attribute__((ext_vector_type(8))) float v8f;

// ---------------- device helpers ----------------

// Upward recurrence, same operation order as reference _sph_jn (f32).
__device__ __forceinline__ float sph_jl(int l, float z) {
  float s = sinf(z), c = cosf(z);
  float j0 = s / z;
  if (l == 0) return j0;
  float j1 = s / (z * z) - c / z;
  float jm = j0, jc = j1;
  for (int i = 1; i < l; ++i) {
    float jn = (2.0f * i + 1.0f) / z * jc - jm;
    jm = jc; jc = jn;
  }
  return jc;
}

// cbf via WMMA: per wave, cbf[l][t16] = Coef(16x8) x Powers(8x16), two
// chained V_WMMA_F32_16X16X4_F32 (K=0..3 then K=4..7). Writes transposed
// rows cbf[t][l] (stride 20) into lds_cbf. All lanes active (EXEC all-1s).
__device__ __forceinline__ void cbf_wmma(const float* lds_ct, const float* lds_coef,
                                         float* lds_cbf, int tid) {
  const int wave = tid >> 5;
  const int lane = tid & 31;
  const int col  = lane & 15;        // = A row M, = B col N
  const bool hi  = lane >= 16;       // lane half selects K pair {0,2|1,3} / {4,6|5,7}
  const int  tl  = wave * 16 + col;

  float ct = lds_ct[tl];
  float c2 = ct * ct, c3 = c2 * ct, c4 = c2 * c2, c5 = c4 * ct, c6 = c4 * c2;

  // B (4x16, KxN): VGPR0 holds K=0(lo)/K=2(hi), VGPR1 holds K=1(lo)/K=3(hi)
  v2f B1 = { hi ? c2 : 1.0f, hi ? c3 : ct };   // powers x^0..x^3
  v2f B2 = { hi ? c6 : c4,   hi ? 0.0f : c5 }; // powers x^4..x^6 (k=7 coef = 0)

  // A (16x4, MxK): same K pairing, M = col
  const float* cr = &lds_coef[col * 8];
  v2f A1 = { cr[hi ? 2 : 0], cr[hi ? 3 : 1] };
  v2f A2 = { cr[hi ? 6 : 4], cr[hi ? 7 : 5] };

  v8f acc = {};
  acc = __builtin_amdgcn_wmma_f32_16x16x4_f32(false, A1, false, B1, (short)0, acc, false, false);
  acc = __builtin_amdgcn_wmma_f32_16x16x4_f32(false, A2, false, B2, (short)0, acc, false, false);

  // D: VGPR r holds (M = r + 8*hi, N = col) -> store cbf[tl][l], l contiguous
  float* row = &lds_cbf[tl * 20 + (hi ? 8 : 0)];
  v4f lo = { acc[0], acc[1], acc[2], acc[3] };
  v4f hh = { acc[4], acc[5], acc[6], acc[7] };
  *(v4f*)(row)     = lo;
  *(v4f*)(row + 4) = hh;
}

// ---------------- phase 1: rbf table (E x 42), fully coalesced ----------------
__global__ __launch_bounds__(256) void rbf_table_kernel(
    const float* __restrict__ dist, float* __restrict__ rbf,
    long long total, Tables tb) {
  __shared__ float zs[NJ], ns[NJ];
  int tid = threadIdx.x;
  if (tid < NJ) { zs[tid] = tb.zeros[tid]; ns[tid] = tb.norms[tid]; }
  __syncthreads();
  long long i = (long long)blockIdx.x * 256 + tid;
  if (i >= total) return;
  int e = (int)(i / NJ);
  int j = (int)(i - (long long)e * NJ);
  int l = j / NR;
  float x = dist[e] * CUTOFF_INV;
  rbf[i] = ns[j] * sph_jl(l, x * zs[j]);
}

// ---------------- phase 2: gather * cbf, coalesced stores ----------------
__global__ __launch_bounds__(256) void combine_kernel(
    const float* __restrict__ angle, const int* __restrict__ idx_kj,
    const float* __restrict__ rbf, float* __restrict__ out,
    int T, Tables tb) {
  __shared__ float lds_ct[TILE_T];
  __shared__ int   lds_idx[TILE_T];
  __shared__ float lds_coef[16 * 8];
  __shared__ __align__(16) float lds_cbf[TILE_T * 20];

  int tid = threadIdx.x;
  int t0  = blockIdx.x * TILE_T;
  if (tid < TILE_T) {
    int t = t0 + tid;
    lds_ct[tid]  = cosf((t < T) ? angle[t] : 0.0f);
    lds_idx[tid] = (t < T) ? idx_kj[t] : 0;
  }
  if (tid < 128) lds_coef[tid] = (&tb.coefA[0][0])[tid];
  __syncthreads();

  cbf_wmma(lds_ct, lds_coef, lds_cbf, tid);
  __syncthreads();

  long long base  = (long long)t0 * NJ;
  long long total = (long long)T * NJ;
#pragma unroll
  for (int s = 0; s < (TILE_T * NJ) / 256; ++s) {   // 21 iterations
    int flat = tid + s * 256;
    long long g = base + flat;
    if (g < total) {
      int tl = flat / NJ;
      int j  = flat - tl * NJ;
      int l  = j / NR;
      float r = rbf[(long long)lds_idx[tl] * NJ + j];
      out[g] = r * lds_cbf[tl * 20 + l];
    }
  }
}

// ---------------- fused fallback (no workspace table) ----------------
__global__ __launch_bounds__(256) void fused_kernel(
    const float* __restrict__ dist, const float* __restrict__ angle,
    const int* __restrict__ idx_kj, float* __restrict__ out,
    int T, Tables tb) {
  __shared__ float lds_ct[TILE_T];
  __shared__ int   lds_idx[TILE_T];
  __shared__ float lds_coef[16 * 8];
  __shared__ float zs[NJ], ns[NJ];
  __shared__ __align__(16) float lds_cbf[TILE_T * 20];

  int tid = threadIdx.x;
  int t0  = blockIdx.x * TILE_T;
  if (tid < TILE_T) {
    int t = t0 + tid;
    lds_ct[tid]  = cosf((t < T) ? angle[t] : 0.0f);
    lds_idx[tid] = (t < T) ? idx_kj[t] : 0;
  }
  if (tid < 128) lds_coef[tid] = (&tb.coefA[0][0])[tid];
  if (tid < NJ) { zs[tid] = tb.zeros[tid]; ns[tid] = tb.norms[tid]; }
  __syncthreads();

  cbf_wmma(lds_ct, lds_coef, lds_cbf, tid);
  __syncthreads();

  long long base  = (long long)t0 * NJ;
  long long total = (long long)T * NJ;
#pragma unroll
  for (int s = 0; s < (TILE_T * NJ) / 256; ++s) {
    int flat = tid + s * 256;
    long long g = base + flat;
    if (g < total) {
      int tl = flat / NJ;
      int j  = flat - tl * NJ;
      int l  = j / NR;
      float x = dist[lds_idx[tl]] * CUTOFF_INV;
      float r = ns[j] * sph_jl(l, x * zs[j]);
      out[g] = r * lds_cbf[tl * 20 + l];
    }
  }
}

// ---------------- host: deterministic table construction ----------------
static double jn_d(int order, double r) {
  double j0 = sin(r) / r;
  if (order == 0) return j0;
  double j1 = sin(r) / (r * r) - cos(r) / r;
  double jm = j0, jc = j1;
  for (int i = 1; i < order; ++i) {
    double jn = (2.0 * i + 1.0) / r * jc - jm;
    jm = jc; jc = jn;
  }
  return jc;
}

static double bisect_root(int order, double a, double b) {
  double fa = jn_d(order, a);
  for (int it = 0; it < 100; ++it) {
    double m = 0.5 * (a + b);
    double fm = jn_d(order, m);
    if (fa * fm > 0.0) { a = m; fa = fm; } else { b = m; }
  }
  return 0.5 * (a + b);
}

static void build_tables(Tables& tb) {
  const int NPTS = NS + NR - 1;   // 12
  double zeros[NS][NR];
  double points[NPTS], racines[NPTS];
  for (int k = 0; k < NR; ++k) zeros[0][k] = (k + 1) * M_PI;
  for (int j = 0; j < NPTS; ++j) points[j] = (j + 1) * M_PI;
  for (int j = 0; j < NPTS; ++j) racines[j] = 0.0;
  for (int i = 1; i < NS; ++i) {
    int cnt = NPTS - i;
    for (int j = 0; j < cnt; ++j) racines[j] = bisect_root(i, points[j], points[j + 1]);
    for (int j = 0; j < NPTS; ++j) points[j] = racines[j];
    for (int k = 0; k < NR; ++k) zeros[i][k] = racines[k];
  }
  for (int l = 0; l < NS; ++l)
    for (int k = 0; k < NR; ++k) {
      tb.zeros[l * NR + k] = (float)zeros[l][k];
      tb.norms[l * NR + k] = (float)(sqrt(2.0) / fabs(jn_d(l + 1, zeros[l][k])));
    }
  // Legendre monomial coefficients P_l(x), l = 0..6
  static const double LC[NS][8] = {
    { 1, 0, 0, 0, 0, 0, 0, 0 },
    { 0, 1, 0, 0, 0, 0, 0, 0 },
    { -0.5, 0, 1.5, 0, 0, 0, 0, 0 },
    { 0, -1.5, 0, 2.5, 0, 0, 0, 0 },
    { 0.375, 0, -3.75, 0, 4.375, 0, 0, 0 },
    { 0, 1.875, 0, -8.75, 0, 7.875, 0, 0 },
    { -0.3125, 0, 6.5625, 0, -19.6875, 0, 14.4375, 0 }
  };
  for (int l = 0; l < 16; ++l)
    for (int k = 0; k < 8; ++k) tb.coefA[l][k] = 0.0f;
  for (int l = 0; l < NS; ++l) {
    double pref = sqrt((2.0 * l + 1.0) / (4.0 * M_PI));
    for (int k = 0; k < 8; ++k) tb.coefA[l][k] = (float)(pref * LC[l][k]);
  }
}

extern "C" void kernel_launch(void* const* d_in, const int* in_sizes, int n_in,
                              void* d_out, int out_size, void* d_ws, size_t ws_size,
                              hipStream_t stream) {
  const float* dist  = (const float*)d_in[0];
  const float* angle = (const float*)d_in[1];
  const int*   idx   = (const int*)d_in[2];
  float*       out   = (float*)d_out;
  const int E = in_sizes[0];
  const int T = in_sizes[1];

  Tables tb;
  build_tables(tb);

  const long long rbf_total = (long long)E * NJ;
  const int blocks2 = (T + TILE_T - 1) / TILE_T;

  if (ws_size >= (size_t)rbf_total * sizeof(float)) {
    float* rbf = (float*)d_ws;
    const int blocks1 = (int)((rbf_total + 255) / 256);
    rbf_table_kernel<<<dim3(blocks1), dim3(256), 0, stream>>>(dist, rbf, rbf_total, tb);
    combine_kernel<<<dim3(blocks2), dim3(256), 0, stream>>>(angle, idx, rbf, out, T, tb);
  } else {
    fused_kernel<<<dim3(blocks2), dim3(256), 0, stream>>>(dist, angle, idx, out, T, tb);
  }
}